// Queues_47175920779355
// MI455X (gfx1250) — compile-verified
//
#include <hip/hip_runtime.h>
#include <hip/hip_bf16.h>
#include <stdint.h>

// ---------------------------------------------------------------------------
// Problem: L=4096 label queues, Q=16 slots, B=4096 samples, F=2048 features.
// Net effect of the reference = per-label row gather:
//   out[l,:] = input_features[winner(l),:]  if a sample won slot sample_idx[l]
//            = queues[l, sample_idx[l], :]  otherwise
// winner(l) = index of the r*-th occurrence of label l, where
//   t  = (sample_idx[l] - flags[l]) mod Q
//   r* = (c-1) - ((c-1-t) mod Q)   (valid iff c>0 and t <= c-1), c = #samples
// Traffic = 32MB read + 32MB write -> ~2.8us roofline @ 23.3 TB/s.
// Phase 2 uses the CDNA5 Tensor Data Mover (tensor_load_to_lds /
// tensor_store_from_lds, TENSORcnt) to DMA each 8KB row, one wave per row.
// This toolchain exposes the 6-arg TDM builtins: (g0 v4u, g1 v8i, g2 v4i,
// g3 v4i, g4 v8i [unused 5th SGPR group -> zeros], cpol i32).
// ---------------------------------------------------------------------------

typedef __attribute__((ext_vector_type(4))) unsigned int v4u;
typedef __attribute__((ext_vector_type(8))) int          v8i;
typedef __attribute__((ext_vector_type(4))) int          v4i;

// ---------------- Phase 0: zero the per-block histogram --------------------
__global__ void zero_ws(int* p, int n) {
    int i = blockIdx.x * blockDim.x + threadIdx.x;
    if (i < n) p[i] = 0;
}

// ---------------- Phase 1a: per-256-sample-block label histogram -----------
__global__ void block_hist(const int* __restrict__ labels, int B, int L,
                           int* __restrict__ blockHist) {
    int b = blockIdx.x;
    int i = b * 256 + threadIdx.x;
    if (i < B) {
        atomicAdd(&blockHist[(size_t)b * L + labels[i]], 1);
    }
}

// ---------------- Phase 1b: closed-form winner per label -------------------
__global__ void winner_kernel(const int* __restrict__ labels,
                              const int* __restrict__ flags,
                              const int* __restrict__ sidx,
                              const int* __restrict__ blockHist,
                              int* __restrict__ winner,
                              int B, int L, int Q, int NB) {
    int l = blockIdx.x * blockDim.x + threadIdx.x;
    if (l >= L) return;

    // total count of label l
    int c = 0;
    for (int b = 0; b < NB; ++b) c += blockHist[(size_t)b * L + l];

    int s  = sidx[l] % Q;
    int fl = flags[l] % Q;
    int t  = s - fl;  if (t < 0) t += Q;   // required rank mod Q

    int w = -1;
    if (c > 0 && t <= c - 1) {
        int r = (c - 1) - ((c - 1 - t) % Q);   // largest qualifying rank
        // find the block holding occurrence #r (0-indexed)
        int acc = 0, b = 0;
        for (; b < NB; ++b) {
            int cb = blockHist[(size_t)b * L + l];
            if (acc + cb > r) break;
            acc += cb;
        }
        int local = r - acc;                   // local occurrence index in block b
        int base  = b * 256;
        int seen  = 0;
        for (int j = 0; j < 256 && base + j < B; ++j) {
            if (labels[base + j] == l) {
                if (seen == local) { w = base + j; break; }
                ++seen;
            }
        }
    }
    winner[l] = w;
}

// ---------------- Phase 2: TDM row copy (src row -> LDS -> out row) --------
#define TDM_LDS_BYTES 8192  // F=2048 floats

__global__ void __launch_bounds__(32)
copy_rows_tdm(const float* __restrict__ feat,
              const float* __restrict__ queues,
              const int* __restrict__ sidx,
              const int* __restrict__ winner,
              float* __restrict__ out,
              int L, int Q, int F) {
    __shared__ __align__(16) unsigned char lds_buf[TDM_LDS_BYTES];
    int l = blockIdx.x;
    int w = winner[l];
    int s = sidx[l] % Q;

    const float* src = (w >= 0) ? (feat + (size_t)w * F)
                                : (queues + ((size_t)l * Q + s) * F);
    float* dst = out + (size_t)l * F;

    size_t bytes = (size_t)F * 4u;
    if (bytes > TDM_LDS_BYTES) {
        // generic fallback (not hit for this shape): plain vector copy
        const float4* s4 = (const float4*)src;
        float4*       d4 = (float4*)dst;
        int n4 = F / 4;
        for (int i = threadIdx.x; i < n4; i += 32) d4[i] = s4[i];
        return;
    }

    // Flat pointer to LDS: low 32 bits are the wave-relative LDS byte offset.
    unsigned lds_addr = (unsigned)(uintptr_t)&lds_buf[0];
    unsigned n        = (unsigned)F;   // elements, data_size = 4 bytes

    // --- D# group 0 (128b): count=1 | lds_addr | global_addr[56:0] | type=2
    auto make_g0 = [&](unsigned long long ga) {
        v4u g0;
        g0.x = 1u;                                          // count=1 (valid), user mode
        g0.y = lds_addr;                                    // lds_addr (bytes)
        g0.z = (unsigned)(ga & 0xFFFFFFFFull);              // global_addr[31:0]
        g0.w = (unsigned)((ga >> 32) & 0x01FFFFFFull)       // global_addr[56:32]
             | 0x80000000u;                                 // type=2 ("image")
        return g0;
    };

    // --- D# group 1 (256b): wg_mask=0, data_size=2 (4B), 1-row tile of n elems
    v8i g1;
    g1[0] = 0x00020000;                                     // data_size=2 (4 bytes)
    g1[1] = (int)((n & 0xFFFFu) << 16);                     // tensor_dim0[15:0]  @63:48
    g1[2] = (int)(((n >> 16) & 0xFFFFu) | (1u << 16));      // tensor_dim0[31:16] | tensor_dim1=1
    g1[3] = (int)((n & 0xFFFFu) << 16);                     // tensor_dim1 hi=0 | tile_dim0=n
    g1[4] = 1;                                              // tile_dim1=1, tile_dim2=0
    g1[5] = (int)n;                                         // tensor_dim0_stride[31:0]
    g1[6] = 0;                                              // stride hi / dim1_stride lo
    g1[7] = 0;
    v4i g2 = {0, 0, 0, 0};                                  // dims 2/3 unused
    v4i g3 = {0, 0, 0, 0};
    v8i g4 = {0, 0, 0, 0, 0, 0, 0, 0};                      // unused 5th group (VADDR4)

    unsigned long long gsrc = (unsigned long long)(uintptr_t)src;
    unsigned long long gdst = (unsigned long long)(uintptr_t)dst;

    // DMA: global row -> LDS, drain TENSORcnt, LDS -> global out row.
    __builtin_amdgcn_tensor_load_to_lds(make_g0(gsrc), g1, g2, g3, g4, 0);
    __builtin_amdgcn_s_wait_tensorcnt(0);
    __builtin_amdgcn_tensor_store_from_lds(make_g0(gdst), g1, g2, g3, g4, 0);
    __builtin_amdgcn_s_wait_tensorcnt(0);
}

// ---------------------------------------------------------------------------
extern "C" void kernel_launch(void* const* d_in, const int* in_sizes, int n_in,
                              void* d_out, int out_size, void* d_ws, size_t ws_size,
                              hipStream_t stream) {
    const float* feat   = (const float*)d_in[0];   // [B,F]
    const float* queues = (const float*)d_in[1];   // [L,Q,F]
    const int*   labels = (const int*)d_in[2];     // [B]
    const int*   flags  = (const int*)d_in[3];     // [L]
    const int*   sidx   = (const int*)d_in[4];     // [L]

    const int B = in_sizes[2];
    const int L = in_sizes[3];
    const int F = (int)((long long)in_sizes[0] / B);
    const int Q = (int)((long long)in_sizes[1] / ((long long)L * F));
    const int NB = (B + 255) / 256;

    int* winner    = (int*)d_ws;          // L ints
    int* blockHist = winner + L;          // NB*L ints

    int zn = NB * L;
    zero_ws<<<(zn + 255) / 256, 256, 0, stream>>>(blockHist, zn);
    block_hist<<<NB, 256, 0, stream>>>(labels, B, L, blockHist);
    winner_kernel<<<(L + 255) / 256, 256, 0, stream>>>(labels, flags, sidx,
                                                       blockHist, winner,
                                                       B, L, Q, NB);
    copy_rows_tdm<<<L, 32, 0, stream>>>(feat, queues, sidx, winner,
                                        (float*)d_out, L, Q, F);
}